// get_adj_26714696581518
// MI455X (gfx1250) — compile-verified
//
#include <hip/hip_runtime.h>
#include <math.h>

typedef __attribute__((ext_vector_type(2))) float v2f;
typedef __attribute__((ext_vector_type(8))) float v8f;
typedef __attribute__((ext_vector_type(4))) unsigned int v4u;
typedef __attribute__((ext_vector_type(4))) int v4i;
typedef __attribute__((ext_vector_type(8))) int v8i;

#define EPSN 1e-6f

__device__ __forceinline__ float eluf(float x) { return x > 0.0f ? x : expm1f(x); }

// ----------------------------------------------------------------------------
// Problem constants: B=32, NC=62, F=256, NE=512, R=16
//   M1 = B*NC = 1984 rows, NTH = NC*F = 15872, NROWS = B*NC*NC = 123008
// ----------------------------------------------------------------------------

// o[r,f] = bias[i,f] + sum_j p[i,j] * x[b,j,f]     (r = b*62+i)
__global__ void __launch_bounds__(256)
k_mix(const float* __restrict__ x, const float* __restrict__ p,
      const float* __restrict__ bias, float* __restrict__ o)
{
    int r = blockIdx.x;           // 0..1983
    int f = threadIdx.x;          // 0..255
    int b = r / 62, i = r % 62;
    const float* xb = x + (long)b * 62 * 256 + f;
    const float* pi = p + i * 62;
    float acc = bias[i * 256 + f];
    #pragma unroll 2
    for (int j = 0; j < 62; ++j) acc += pi[j] * xb[(long)j * 256];
    o[(long)r * 256 + f] = acc;
}

// ----------------------------------------------------------------------------
// Generic fp32 WMMA GEMM: C[M,N] = op(A[M,K] @ B[K,N]), row-major.
// Each wave owns a 32x64 tile: 2 M-subtiles x 4 N-subtiles, so each B
// fragment is amortized over two WMMAs (halves B traffic from L2).
// Fragment rule (V_WMMA_F32_16X16X4_F32):  K = k0 + 2*half + vgpr
//   A: lane m=lane&15 holds A[m, k0+2h], A[m, k0+2h+1]   (contiguous float2)
//   B: lane n=lane&15 holds B[k0+2h, n], B[k0+2h+1, n]
//   C/D: vgpr v, lane -> (M = v + 8*half, N = lane&15)
// ----------------------------------------------------------------------------
template <bool ELU_ACT>
__global__ void __launch_bounds__(256)
gemm_wmma_f32(const float* __restrict__ A, const float* __restrict__ B,
              float* __restrict__ C, int tilesM32, int tilesN64,
              int lda, int ldb, int ldc, int K)
{
    int wave = (blockIdx.x * blockDim.x + threadIdx.x) >> 5;
    int lane = threadIdx.x & 31;
    int tm = wave / tilesN64;
    int tn = wave % tilesN64;
    if (tm >= tilesM32) return;
    int half = lane >> 4;
    int mr   = lane & 15;

    const float* Arow0 = A + (long)(tm * 32 + mr) * lda;
    const float* Arow1 = Arow0 + (long)16 * lda;
    const float* Bcol  = B + (long)tn * 64 + mr;

    v8f acc[2][4] = {};
    for (int k = 0; k < K; k += 4) {
        int ka = k + 2 * half;
        v2f a0 = *reinterpret_cast<const v2f*>(Arow0 + ka);
        v2f a1 = *reinterpret_cast<const v2f*>(Arow1 + ka);
        const float* Bk = Bcol + (long)ka * ldb;
        __builtin_prefetch(Bk + (long)8 * ldb, 0, 3);   // global_prefetch_b8
        v2f b0, b1, b2, b3;
        b0.x = Bk[0];  b0.y = Bk[ldb + 0];
        b1.x = Bk[16]; b1.y = Bk[ldb + 16];
        b2.x = Bk[32]; b2.y = Bk[ldb + 32];
        b3.x = Bk[48]; b3.y = Bk[ldb + 48];
        acc[0][0] = __builtin_amdgcn_wmma_f32_16x16x4_f32(false, a0, false, b0, (short)0, acc[0][0], false, false);
        acc[0][1] = __builtin_amdgcn_wmma_f32_16x16x4_f32(false, a0, false, b1, (short)0, acc[0][1], false, false);
        acc[0][2] = __builtin_amdgcn_wmma_f32_16x16x4_f32(false, a0, false, b2, (short)0, acc[0][2], false, false);
        acc[0][3] = __builtin_amdgcn_wmma_f32_16x16x4_f32(false, a0, false, b3, (short)0, acc[0][3], false, false);
        acc[1][0] = __builtin_amdgcn_wmma_f32_16x16x4_f32(false, a1, false, b0, (short)0, acc[1][0], false, false);
        acc[1][1] = __builtin_amdgcn_wmma_f32_16x16x4_f32(false, a1, false, b1, (short)0, acc[1][1], false, false);
        acc[1][2] = __builtin_amdgcn_wmma_f32_16x16x4_f32(false, a1, false, b2, (short)0, acc[1][2], false, false);
        acc[1][3] = __builtin_amdgcn_wmma_f32_16x16x4_f32(false, a1, false, b3, (short)0, acc[1][3], false, false);
    }
    #pragma unroll
    for (int mi = 0; mi < 2; ++mi) {
        #pragma unroll
        for (int v = 0; v < 8; ++v) {
            float* cp = C + (long)(tm * 32 + mi * 16 + v + half * 8) * ldc
                          + (long)tn * 64 + mr;
            #pragma unroll
            for (int s = 0; s < 4; ++s) {
                float xv = acc[mi][s][v];
                if (ELU_ACT) xv = eluf(xv);
                cp[16 * s] = xv;
            }
        }
    }
}

// denom[r,d] = eps + sum_{j<62} |g[r, j*256+d]|
__global__ void __launch_bounds__(256)
k_denom(const float* __restrict__ g, float* __restrict__ den)
{
    int r = blockIdx.x, d = threadIdx.x;
    const float* gr = g + (long)r * 15872 + d;
    float acc = 0.f;
    #pragma unroll 2
    for (int j = 0; j < 62; ++j) acc += fabsf(gr[j * 256]);
    den[(long)r * 256 + d] = acc + EPSN;
}

// z[n,d] = g[n,d] / denom[n/62, d]   (in place; g flat == z layout)
__global__ void __launch_bounds__(256)
k_normz(float* __restrict__ z, const float* __restrict__ den)
{
    long n = blockIdx.x;
    int d = threadIdx.x;
    int r = (int)(n / 62);
    z[n * 256 + d] = z[n * 256 + d] / den[(long)r * 256 + d];
}

// |codebook[c]|^2
__global__ void __launch_bounds__(256)
k_cnorm(const float* __restrict__ cb, float* __restrict__ cn)
{
    int c = blockIdx.x * 256 + threadIdx.x;   // 0..511
    const float* row = cb + (long)c * 256;
    float acc = 0.f;
    for (int k = 0; k < 256; ++k) acc += row[k] * row[k];
    cn[c] = acc;
}

// ----------------------------------------------------------------------------
// VQ: fused  dist = |c|^2 - 2 * (z @ c)  +  argmin over 512 codes.
// Block = 128 thr (4 waves); stage 64 z-rows (64 KB) into LDS via the Tensor
// Data Mover (one wave issues TENSOR_LOAD_TO_LDS, waits TENSORcnt, barrier);
// wave w handles rows [w*16, w*16+16), sweeps 32 code-tiles of 16 with WMMA.
// ----------------------------------------------------------------------------
__global__ void __launch_bounds__(128)
k_vq_argmin(const float* __restrict__ z, const float* __restrict__ cb,
            const float* __restrict__ cn, int* __restrict__ idxOut)
{
    __shared__ float sA[64 * 256];    // 64 KB
    int tid = threadIdx.x;
    long row0 = (long)blockIdx.x * 64;

#if __has_builtin(__builtin_amdgcn_tensor_load_to_lds)
    if (tid < 32) {   // wave 0 issues the TDM descriptor (per-wave instruction)
        unsigned long long ga = (unsigned long long)(const void*)(z + row0 * 256);
        // LDS aperture: low 32 bits of the flat pointer are the LDS byte offset
        unsigned ldsb = (unsigned)(unsigned long long)(void*)sA;
        // D# group0: count=1 | lds_addr | global_addr[56:0] | type=2
        v4u g0 = { 1u, ldsb, (unsigned)ga,
                   (unsigned)((ga >> 32) & 0x01FFFFFFull) | (2u << 30) };
        // D# group1: data_size=4B; tensor_dim0=tile_dim0=16384; dims1=1; 1 row
        v8i g1 = { 0x00020000,   // wg_mask=0 | data_size=2 (4B)
                   0x40000000,   // atomic_bar=0 | tensor_dim0.lo16=0x4000
                   0x00010000,   // tensor_dim0.hi16=0 | tensor_dim1.lo16=1
                   0x40000000,   // tensor_dim1.hi16=0 | tile_dim0=0x4000
                   0x00000001,   // tile_dim1=1 | tile_dim2=0
                   16384,        // tensor_dim0_stride.lo32
                   0, 0 };
        v4i gz = { 0, 0, 0, 0 };
#if defined(__clang_major__) && (__clang_major__ >= 23)
        v8i gz8 = { 0, 0, 0, 0, 0, 0, 0, 0 };
        __builtin_amdgcn_tensor_load_to_lds(g0, g1, gz, gz, gz8, 0);
#else
        __builtin_amdgcn_tensor_load_to_lds(g0, g1, gz, gz, 0);
#endif
        __builtin_amdgcn_s_wait_tensorcnt(0);
    }
#else
    {
        const float4* gsrc = reinterpret_cast<const float4*>(z + row0 * 256);
        float4* ldst = reinterpret_cast<float4*>(sA);
        #pragma unroll 4
        for (int t = 0; t < 128; ++t) ldst[tid + t * 128] = gsrc[tid + t * 128];
    }
#endif
    __syncthreads();

    int wave = tid >> 5, lane = tid & 31;
    int half = lane >> 4, mr = lane & 15;
    const float* Az = sA + (wave * 16 + mr) * 256;

    float minv[8]; int mini[8];
    #pragma unroll
    for (int v = 0; v < 8; ++v) { minv[v] = 3.4e38f; mini[v] = 0; }

    for (int ct = 0; ct < 32; ++ct) {            // 512 codes / 16
        int code = ct * 16 + mr;                  // lane's code (N index)
        const float* crow = cb + (long)code * 256;
        v8f acc = {};
        for (int k = 0; k < 256; k += 4) {
            int ka = k + 2 * half;
            v2f a = *reinterpret_cast<const v2f*>(Az + ka);
            v2f b = *reinterpret_cast<const v2f*>(crow + ka);   // B[k,n]=cb[n,k]
            acc = __builtin_amdgcn_wmma_f32_16x16x4_f32(false, a, false, b, (short)0, acc, false, false);
        }
        float cnv = cn[code];
        #pragma unroll
        for (int v = 0; v < 8; ++v) {
            float dist = cnv - 2.0f * acc[v];
            if (dist < minv[v] || (dist == minv[v] && code < mini[v])) {
                minv[v] = dist; mini[v] = code;
            }
        }
    }
    // min-reduce across the 16 lanes that share each output row
    #pragma unroll
    for (int off = 1; off < 16; off <<= 1) {
        #pragma unroll
        for (int v = 0; v < 8; ++v) {
            float ov = __shfl_xor(minv[v], off, 32);
            int   oi = __shfl_xor(mini[v], off, 32);
            if (ov < minv[v] || (ov == minv[v] && oi < mini[v])) { minv[v] = ov; mini[v] = oi; }
        }
    }
    if (mr == 0) {
        long rb = row0 + wave * 16 + half * 8;
        #pragma unroll
        for (int v = 0; v < 8; ++v) idxOut[rb + v] = mini[v];
    }
}

// per-row partial sums of (zq - z)^2  (deterministic)
__global__ void __launch_bounds__(256)
k_msepart(const float* __restrict__ z, const float* __restrict__ cb,
          const int* __restrict__ idx, float* __restrict__ part)
{
    __shared__ float red[256];
    long n = blockIdx.x;
    int d = threadIdx.x;
    float df = cb[(long)idx[n] * 256 + d] - z[n * 256 + d];
    red[d] = df * df; __syncthreads();
    for (int s = 128; s > 0; s >>= 1) { if (d < s) red[d] += red[d + s]; __syncthreads(); }
    if (d == 0) part[n] = red[0];
}

__global__ void __launch_bounds__(256)
k_reduce_mse(const float* __restrict__ part, float* __restrict__ mse)
{
    __shared__ float red[256];
    float acc = 0.f;
    for (long n = threadIdx.x; n < 123008; n += 256) acc += part[n];
    red[threadIdx.x] = acc; __syncthreads();
    for (int s = 128; s > 0; s >>= 1) { if (threadIdx.x < s) red[threadIdx.x] += red[threadIdx.x + s]; __syncthreads(); }
    if (threadIdx.x == 0) mse[0] = red[0];
}

__global__ void __launch_bounds__(256)
k_hist(const int* __restrict__ idx, int* __restrict__ cnt, int* __restrict__ cntB)
{
    long n = (long)blockIdx.x * 256 + threadIdx.x;   // 481*256 = 123136 threads, no bound check
    int c = idx[n];
    int b = (int)(n / 3844);
    atomicAdd(&cnt[c], 1);
    atomicAdd(&cntB[b * 512 + c], 1);
}

// m[b,f] = (1/3844) * sum_c cntB[b,c] * cb[c,f]
__global__ void __launch_bounds__(256)
k_squeeze(const int* __restrict__ cntB, const float* __restrict__ cb, float* __restrict__ m)
{
    int b = blockIdx.x, f = threadIdx.x;
    const int* cB = cntB + b * 512;
    float acc = 0.f;
    for (int c = 0; c < 512; ++c) acc += (float)cB[c] * cb[(long)c * 256 + f];
    m[b * 256 + f] = acc * (1.0f / 3844.0f);
}

__global__ void __launch_bounds__(512)
k_se1(const float* __restrict__ m, const float* __restrict__ w1,
      const float* __restrict__ b1, float* __restrict__ h)
{
    int t = threadIdx.x;            // 512 = 32*16
    int b = t >> 4, r = t & 15;
    float acc = b1[r];
    for (int f = 0; f < 256; ++f) acc += m[b * 256 + f] * w1[f * 16 + r];
    h[t] = fmaxf(acc, 0.0f);
}

__global__ void __launch_bounds__(256)
k_se2(const float* __restrict__ h, const float* __restrict__ w2,
      const float* __restrict__ b2, float* __restrict__ s)
{
    int b = blockIdx.x, f = threadIdx.x;
    float acc = b2[f];
    #pragma unroll
    for (int r = 0; r < 16; ++r) acc += h[b * 16 + r] * w2[r * 256 + f];
    s[b * 256 + f] = 1.0f / (1.0f + expf(-acc));
}

// u[n] = elu( sum_d elu( cb[idx[n],d] * (1 + s[b,d]) ) )
__global__ void __launch_bounds__(256)
k_final1(const int* __restrict__ idx, const float* __restrict__ cb,
         const float* __restrict__ s, float* __restrict__ u)
{
    __shared__ float red[256];
    long n = blockIdx.x;
    int d = threadIdx.x;
    int b = (int)(n / 3844);
    float v = cb[(long)idx[n] * 256 + d] * (1.0f + s[b * 256 + d]);
    red[d] = eluf(v); __syncthreads();
    for (int st = 128; st > 0; st >>= 1) { if (d < st) red[d] += red[d + st]; __syncthreads(); }
    if (d == 0) u[n] = eluf(red[0]);
}

// out[b,i,j] = u / (sum_j |u| + eps)
__global__ void __launch_bounds__(32)
k_rownorm(const float* __restrict__ u, float* __restrict__ out)
{
    int r = blockIdx.x, j = threadIdx.x;
    float a0 = (j < 62)      ? u[(long)r * 62 + j]      : 0.f;
    float a1 = (j + 32 < 62) ? u[(long)r * 62 + j + 32] : 0.f;
    float sm = fabsf(a0) + fabsf(a1);
    #pragma unroll
    for (int off = 16; off > 0; off >>= 1) sm += __shfl_xor(sm, off, 32);
    float inv = 1.0f / (sm + EPSN);
    if (j < 62)      out[(long)r * 62 + j]      = a0 * inv;
    if (j + 32 < 62) out[(long)r * 62 + j + 32] = a1 * inv;
}

__global__ void __launch_bounds__(256)
k_usage(const int* __restrict__ cnt, float* __restrict__ outUsage)
{
    __shared__ float red[256];
    float acc = 0.f;
    for (int c = threadIdx.x; c < 512; c += 256) {
        float pb = (float)cnt[c] * (1.0f / 123008.0f);
        acc += pb * logf(pb + 1e-10f);
    }
    red[threadIdx.x] = acc; __syncthreads();
    for (int s = 128; s > 0; s >>= 1) { if (threadIdx.x < s) red[threadIdx.x] += red[threadIdx.x + s]; __syncthreads(); }
    if (threadIdx.x == 0) *outUsage = expf(-red[0]);
}

__global__ void __launch_bounds__(32)
k_vqloss(const float* __restrict__ mse, float* __restrict__ out0)
{
    if (threadIdx.x == 0) out0[0] = 1.25f * mse[0] * (1.0f / 31490048.0f);
}

// ----------------------------------------------------------------------------

extern "C" void kernel_launch(void* const* d_in, const int* in_sizes, int n_in,
                              void* d_out, int out_size, void* d_ws, size_t ws_size,
                              hipStream_t stream) {
    const float* x     = (const float*)d_in[0];
    const float* p     = (const float*)d_in[1];
    const float* bias  = (const float*)d_in[2];
    const float* q_mat = (const float*)d_in[3];
    const float* theta = (const float*)d_in[4];
    const float* cb    = (const float*)d_in[5];
    const float* se_w1 = (const float*)d_in[6];
    const float* se_b1 = (const float*)d_in[7];
    const float* se_w2 = (const float*)d_in[8];
    const float* se_b2 = (const float*)d_in[9];
    float* out = (float*)d_out;

    float* ws = (float*)d_ws;
    constexpr long OFF_O    = 0;
    constexpr long OFF_OQ   = OFF_O    + 1984L * 256;
    constexpr long OFF_G    = OFF_OQ   + 1984L * 256;       // g / z (in place)
    constexpr long OFF_DEN  = OFF_G    + 1984L * 15872;
    constexpr long OFF_U    = OFF_DEN  + 1984L * 256;       // reused: mse partials, then u
    constexpr long OFF_M    = OFF_U    + 123008L;
    constexpr long OFF_H    = OFF_M    + 32L * 256;
    constexpr long OFF_S    = OFF_H    + 512L;
    constexpr long OFF_CN   = OFF_S    + 32L * 256;
    constexpr long OFF_IDX  = OFF_CN   + 512L;
    constexpr long OFF_CNT  = OFF_IDX  + 123008L;
    constexpr long OFF_CNTB = OFF_CNT  + 512L;
    constexpr long OFF_MSE  = OFF_CNTB + 32L * 512;

    float* o    = ws + OFF_O;
    float* oq   = ws + OFF_OQ;
    float* g    = ws + OFF_G;     // also z after k_normz
    float* den  = ws + OFF_DEN;
    float* u    = ws + OFF_U;
    float* mvec = ws + OFF_M;
    float* h    = ws + OFF_H;
    float* s    = ws + OFF_S;
    float* cn   = ws + OFF_CN;
    int*   idx  = (int*)(ws + OFF_IDX);
    int*   cnt  = (int*)(ws + OFF_CNT);
    int*   cntB = (int*)(ws + OFF_CNTB);
    float* mse  = ws + OFF_MSE;

    // zero histograms + mse accumulator (contiguous region)
    hipMemsetAsync((void*)cnt, 0, (512 + 32 * 512 + 64) * sizeof(float), stream);

    // 1) o = p @ x + bias
    k_mix<<<1984, 256, 0, stream>>>(x, p, bias, o);

    // 2) oq = o @ q_mat         (M=1984, N=256,   K=256)  tiles: 62 x 4
    {
        int waves = 62 * 4;
        gemm_wmma_f32<false><<<(waves * 32 + 255) / 256, 256, 0, stream>>>(
            o, q_mat, oq, 62, 4, 256, 256, 256, 256);
    }
    // 3) g = elu(oq @ theta)    (M=1984, N=15872, K=256)  tiles: 62 x 248
    {
        int waves = 62 * 248;
        gemm_wmma_f32<true><<<(waves * 32 + 255) / 256, 256, 0, stream>>>(
            oq, theta, g, 62, 248, 256, 15872, 15872, 256);
    }

    // 4) L1 row denominators + normalize (g -> z, in place)
    k_denom<<<1984, 256, 0, stream>>>(g, den);
    k_normz<<<123008, 256, 0, stream>>>(g, den);

    // 5) VQ: |c|^2, then fused WMMA distance + argmin (TDM-staged LDS tiles)
    k_cnorm<<<2, 256, 0, stream>>>(cb, cn);
    k_vq_argmin<<<1922, 128, 0, stream>>>(g, cb, cn, idx);

    // 6) vq_loss = 1.25 * mean((zq - z)^2)   (deterministic two-stage)
    k_msepart<<<123008, 256, 0, stream>>>(g, cb, idx, u);
    k_reduce_mse<<<1, 256, 0, stream>>>(u, mse);
    k_vqloss<<<1, 32, 0, stream>>>(mse, out);                       // out[0]

    // 7) histograms -> usage (perplexity) and SE squeeze input
    k_hist<<<481, 256, 0, stream>>>(idx, cnt, cntB);
    k_usage<<<1, 256, 0, stream>>>(cnt, out + 1 + 123008);          // out[123009]

    // 8) SE block: s = sigmoid(relu(m@W1+b1)@W2+b2)
    k_squeeze<<<32, 256, 0, stream>>>(cntB, cb, mvec);
    k_se1<<<1, 512, 0, stream>>>(mvec, se_w1, se_b1, h);
    k_se2<<<32, 256, 0, stream>>>(h, se_w2, se_b2, s);

    // 9) u[n] = elu(sum_d elu(zq*(1+s)));  out_adj = u / (L1 row sum + eps)
    k_final1<<<123008, 256, 0, stream>>>(idx, cb, s, u);
    k_rownorm<<<1984, 32, 0, stream>>>(u, out + 1);                 // out[1..123008]

    // 10) codebook passthrough                                      out[123010..]
    hipMemcpyAsync(out + 1 + 123008 + 1, cb, 512L * 256 * sizeof(float),
                   hipMemcpyDeviceToDevice, stream);
}